// DDGCRN_61692910239933
// MI455X (gfx1250) — compile-verified
//
#include <hip/hip_runtime.h>

#define Bq 32
#define Tq 12
#define Nq 883
#define Dq 10
#define Hq 64
#define XI 65      // dim_in = 1 + H
#define NTILES 56  // ceil(883/16)
#define NP 896     // padded node count (56*16)
#define XP 80      // padded xin/yagg row stride (5*16)
#define DP 12      // padded nodevec row stride

typedef __attribute__((ext_vector_type(2))) float v2f;
typedef __attribute__((ext_vector_type(8))) float v8f;

__device__ __forceinline__ float sigf(float x) { return 1.0f / (1.0f + expf(-x)); }

// ---------------------------------------------------------------- fill
__global__ void fill_kernel(float* p, int n, float v) {
    int i = blockIdx.x * blockDim.x + threadIdx.x;
    if (i < n) p[i] = v;
}

// ---------------------------------------------------------------- e1 = ne1 * tid_emb[idx] * diw_emb[idx]
__global__ void embed_kernel(const float* __restrict__ src, const float* __restrict__ ne1,
                             const float* __restrict__ temb, const float* __restrict__ demb,
                             float* __restrict__ e1) {
    int idx = blockIdx.x * blockDim.x + threadIdx.x;   // over B*T*N
    if (idx >= Bq * Tq * Nq) return;
    int n = idx % Nq;
    const float* s = src + (size_t)idx * 3;
    int ti = (int)(s[1] * 288.0f); if (ti < 0) ti = 0; if (ti > 287) ti = 287;
    int di = (int)(s[2]);          if (di < 0) di = 0; if (di > 6) di = 6;
    float* o = e1 + (size_t)idx * Dq;
    #pragma unroll
    for (int d = 0; d < Dq; ++d)
        o[d] = ne1[n * Dq + d] * temb[ti * Dq + d] * demb[di * Dq + d];
}

// ---------------------------------------------------------------- xin = concat([x_t, (z?)*state])  (padded layout)
__global__ void build_xin_kernel(const float* __restrict__ xsrc, int xstride, int t,
                                 const float* __restrict__ state, const float* __restrict__ zr,
                                 float* __restrict__ xin) {
    int idx = blockIdx.x * blockDim.x + threadIdx.x;   // over B*N (valid rows)
    if (idx >= Bq * Nq) return;
    int b = idx / Nq, n = idx % Nq;
    float* o = xin + ((size_t)b * NP + n) * XP;
    o[0] = xsrc[(((size_t)b * Tq + t) * Nq + n) * xstride];
    const float* st = state + (size_t)idx * Hq;
    if (zr) {
        const float* z = zr + (size_t)idx * 128;       // z = zr[..., :64]
        #pragma unroll 8
        for (int h = 0; h < Hq; ++h) o[1 + h] = z[h] * st[h];
    } else {
        #pragma unroll 8
        for (int h = 0; h < Hq; ++h) o[1 + h] = st[h];
    }
}

// ---------------------------------------------------------------- hyper-MLP -> nodevec = tanh(e1_t * filt)
__global__ void mlp_nodevec_kernel(const float* __restrict__ xin, const float* __restrict__ e1, int t,
                                   const float* __restrict__ f1w, const float* __restrict__ f1b,
                                   const float* __restrict__ f2w, const float* __restrict__ f2b,
                                   const float* __restrict__ f3w, const float* __restrict__ f3b,
                                   float* __restrict__ nv) {
    int idx = blockIdx.x * blockDim.x + threadIdx.x;   // over B*N
    if (idx >= Bq * Nq) return;
    int b = idx / Nq, n = idx % Nq;
    float xv[XI];
    const float* xp = xin + ((size_t)b * NP + n) * XP;
    #pragma unroll 5
    for (int i = 0; i < XI; ++i) xv[i] = xp[i];
    float h1[16];
    for (int j = 0; j < 16; ++j) {
        float s = f1b[j];
        #pragma unroll 5
        for (int i = 0; i < XI; ++i) s += f1w[j * XI + i] * xv[i];
        h1[j] = sigf(s);
    }
    float h2[2];
    #pragma unroll
    for (int j = 0; j < 2; ++j) {
        float s = f2b[j];
        #pragma unroll
        for (int i = 0; i < 16; ++i) s += f2w[j * 16 + i] * h1[i];
        h2[j] = sigf(s);
    }
    const float* ep = e1 + (((size_t)b * Tq + t) * Nq + n) * Dq;
    float* op = nv + ((size_t)b * NP + n) * DP;
    #pragma unroll
    for (int d = 0; d < Dq; ++d) {
        float filt = f3b[d] + f3w[d * 2 + 0] * h2[0] + f3w[d * 2 + 1] * h2[1];
        op[d] = tanhf(ep[d] * filt);
    }
    op[10] = 0.0f; op[11] = 0.0f;        // keep K padding zero
}

// ---------------------------------------------------------------- dinv[b,n] = (1 + sum_m relu(v_n . v_m))^-1/2
__global__ void dinv_kernel(const float* __restrict__ nv, float* __restrict__ dinv) {
    __shared__ float snv[Nq * DP];       // 42.4 KB
    int b = blockIdx.x;
    int tid = threadIdx.x;
    const float* nb = nv + (size_t)b * NP * DP;
    for (int i = tid; i < Nq * DP; i += 256) snv[i] = nb[i];
    __syncthreads();
    int n = blockIdx.y * 256 + tid;
    if (n >= Nq) return;
    float v[Dq];
    #pragma unroll
    for (int d = 0; d < Dq; ++d) v[d] = snv[n * DP + d];
    float s = 1.0f;                      // identity contribution
    for (int m = 0; m < Nq; ++m) {
        float acc = 0.0f;
        #pragma unroll
        for (int d = 0; d < Dq; ++d) acc += v[d] * snv[m * DP + d];
        s += fmaxf(acc, 0.0f);
    }
    dinv[b * NP + n] = rsqrtf(s);
}

// ---------------------------------------------------------------- fused WMMA:  y = D*relu(v v^T)*D*xin + d^2 * xin
// one wave per (b, 16-row n-tile); stream 16-col m-tiles; x tiles staged via async global->LDS DMA
__global__ __launch_bounds__(32) void lap_agg_kernel(const float* __restrict__ nv,
                                                     const float* __restrict__ dinv,
                                                     const float* __restrict__ xin,
                                                     float* __restrict__ y) {
    __shared__ float  lds_a[16 * 17];
    __shared__ float4 lds_x4[16 * XP / 4];           // 16 rows x 80 cols
    float* lds_x = (float*)lds_x4;
    int ntile = blockIdx.x, b = blockIdx.y;
    int n0 = ntile * 16;
    int lane = threadIdx.x;
    int lo = lane & 15, hi = lane >> 4;

    const float* nvb = nv + (size_t)b * NP * DP;
    const float* xb  = xin + (size_t)b * NP * XP;
    const float* db  = dinv + (size_t)b * NP;

    // A operand (rows n0..n0+15, K=12 zero-padded): lane M = lo, chunk c holds K = 4c + 2*hi + {0,1}
    v2f aop[3];
    int arow = n0 + lo;
    #pragma unroll
    for (int c = 0; c < 3; ++c)
        aop[c] = *(const v2f*)(nvb + arow * DP + 4 * c + 2 * hi);

    v8f yacc[5];
    #pragma unroll
    for (int ct = 0; ct < 5; ++ct) yacc[ct] = (v8f){};

    unsigned ldsx_base = (unsigned)(uintptr_t)lds_x;   // low 32 bits of flat addr = LDS offset

    for (int mt = 0; mt < NTILES; ++mt) {
        int m0 = mt * 16;
        int brow = m0 + lo;

        // previous iteration's LDS reads must land before async writes may overwrite lds_x
        asm volatile("s_wait_dscnt 0x0" ::: "memory");

        // async DMA: contiguous 16x80 x-tile directly into LDS (10 b128 per lane)
        const float4* xt = (const float4*)(xb + (size_t)m0 * XP);
        #pragma unroll
        for (int i = 0; i < 10; ++i) {
            unsigned loff = ldsx_base + (unsigned)((lane + 32 * i) * 16);
            const float4* gp = xt + lane + 32 * i;
            asm volatile("global_load_async_to_lds_b128 %0, %1, off"
                         :: "v"(loff), "v"(gp) : "memory");
        }

        // B operand = nodevec^T tile: lane N = lo, chunk K mapping mirrors A
        v2f bop[3];
        #pragma unroll
        for (int c = 0; c < 3; ++c)
            bop[c] = *(const v2f*)(nvb + brow * DP + 4 * c + 2 * hi);
        float dcol = db[brow];

        v8f acc = (v8f){};
        acc = __builtin_amdgcn_wmma_f32_16x16x4_f32(false, aop[0], false, bop[0], (short)0, acc, false, false);
        acc = __builtin_amdgcn_wmma_f32_16x16x4_f32(false, aop[1], false, bop[1], (short)0, acc, false, false);
        acc = __builtin_amdgcn_wmma_f32_16x16x4_f32(false, aop[2], false, bop[2], (short)0, acc, false, false);

        // relu + column (d_m) scaling; spill to LDS for C->A layout flip (identity handled in epilogue)
        #pragma unroll
        for (int j = 0; j < 8; ++j) {
            int M = j + 8 * hi;                        // C/D layout row
            lds_a[M * 17 + lo] = fmaxf(acc[j], 0.0f) * dcol;
        }

        // second GEMM: y(16x80) += A'(16x16) @ X(16x80), K=16 in four 4-chunks
        // (DS ops are in-order within a wave, so the lds_a store->read flip needs no wait)
        v2f a2[4];
        #pragma unroll
        for (int kc = 0; kc < 4; ++kc) {
            int k = 4 * kc + 2 * hi;
            a2[kc] = (v2f){lds_a[lo * 17 + k], lds_a[lo * 17 + k + 1]};
        }

        asm volatile("s_wait_asynccnt 0x0" ::: "memory");   // x tile landed in LDS

        #pragma unroll
        for (int ct = 0; ct < 5; ++ct) {
            int cc = ct * 16 + lo;
            #pragma unroll
            for (int kc = 0; kc < 4; ++kc) {
                int kr = 4 * kc + 2 * hi;
                v2f bx = (v2f){lds_x[kr * XP + cc], lds_x[(kr + 1) * XP + cc]};
                yacc[ct] = __builtin_amdgcn_wmma_f32_16x16x4_f32(false, a2[kc], false, bx, (short)0, yacc[ct], false, false);
            }
        }
    }

    // epilogue: y = yacc * d_n + d_n^2 * x[n,:]   (identity term of D(A+I)D x); padded rows have d=0
    #pragma unroll
    for (int j = 0; j < 8; ++j) {
        int grow = n0 + j + 8 * hi;
        float dr = db[grow];
        float dr2 = dr * dr;
        #pragma unroll
        for (int ct = 0; ct < 5; ++ct) {
            int c = ct * 16 + lo;
            float xv = xb[(size_t)grow * XP + c];
            y[((size_t)b * NP + grow) * XP + c] = yacc[ct][j] * dr + dr2 * xv;
        }
    }
}

// ---------------------------------------------------------------- per-node projection: out = act(xg . (e2@wp) + e2@bp)
__global__ void proj_kernel(const float* __restrict__ xin, const float* __restrict__ yagg,
                            const float* __restrict__ e2, const float* __restrict__ wp,
                            const float* __restrict__ bp, int dout, int act,
                            float* __restrict__ out) {
    __shared__ float wsh[130 * 64];
    __shared__ float bsh[64];
    __shared__ float xv[130];
    int n = blockIdx.x;
    int o0 = blockIdx.y * 64;
    int tid = threadIdx.x;

    float e[Dq];
    #pragma unroll
    for (int d = 0; d < Dq; ++d) e[d] = e2[n * Dq + d];

    for (int idx = tid; idx < 130 * 64; idx += 128) {
        int i = idx >> 6, oc = idx & 63;
        int k = i / XI, ii = i - k * XI;
        int o = o0 + oc;
        float s = 0.0f;
        #pragma unroll
        for (int d = 0; d < Dq; ++d) s += e[d] * wp[(((size_t)d * 2 + k) * XI + ii) * dout + o];
        wsh[idx] = s;
    }
    for (int oc = tid; oc < 64; oc += 128) {
        float s = 0.0f;
        #pragma unroll
        for (int d = 0; d < Dq; ++d) s += e[d] * bp[d * dout + o0 + oc];
        bsh[oc] = s;
    }
    __syncthreads();

    for (int b = 0; b < Bq; ++b) {
        const float* xr = xin  + ((size_t)b * NP + n) * XP;
        const float* yr = yagg + ((size_t)b * NP + n) * XP;
        for (int i = tid; i < XI; i += 128) { xv[i] = xr[i]; xv[XI + i] = yr[i]; }
        __syncthreads();
        for (int oc = tid; oc < 64; oc += 128) {
            float s = bsh[oc];
            for (int i = 0; i < 130; ++i) s += xv[i] * wsh[i * 64 + oc];
            s = act ? tanhf(s) : sigf(s);
            out[((size_t)b * Nq + n) * dout + o0 + oc] = s;
        }
        __syncthreads();
    }
}

// ---------------------------------------------------------------- state = r*state + (1-r)*hc
__global__ void state_update_kernel(float* __restrict__ state, const float* __restrict__ zr,
                                    const float* __restrict__ hc) {
    int idx = blockIdx.x * blockDim.x + threadIdx.x;   // over B*N*H
    if (idx >= Bq * Nq * Hq) return;
    int bn = idx >> 6, h = idx & 63;
    float r = zr[(size_t)bn * 128 + 64 + h];
    state[idx] = r * state[idx] + (1.0f - r) * hc[idx];
}

// ---------------------------------------------------------------- end convs
__global__ void endconv1_kernel(const float* __restrict__ h, const float* __restrict__ c1w,
                                const float* __restrict__ c1b, const float* __restrict__ c2w,
                                const float* __restrict__ c2b, const float* __restrict__ src,
                                float* __restrict__ out, float* __restrict__ src2) {
    int idx = blockIdx.x * blockDim.x + threadIdx.x;   // over B*12*N
    if (idx >= Bq * Tq * Nq) return;
    int b = idx / (Tq * Nq), r = idx % (Tq * Nq), p = r / Nq, n = r % Nq;
    const float* hr = h + ((size_t)b * Nq + n) * Hq;
    float s1 = c1b[p], s2 = c2b[p];
    #pragma unroll 8
    for (int k = 0; k < Hq; ++k) { float hv = hr[k]; s1 += hv * c1w[p * Hq + k]; s2 += hv * c2w[p * Hq + k]; }
    out[idx]  = s1;
    src2[idx] = src[(size_t)idx * 3] - s2;
}

__global__ void endconv2_kernel(const float* __restrict__ h, const float* __restrict__ c3w,
                                const float* __restrict__ c3b, float* __restrict__ out) {
    int idx = blockIdx.x * blockDim.x + threadIdx.x;
    if (idx >= Bq * Tq * Nq) return;
    int b = idx / (Tq * Nq), r = idx % (Tq * Nq), p = r / Nq, n = r % Nq;
    const float* hr = h + ((size_t)b * Nq + n) * Hq;
    float s = c3b[p];
    #pragma unroll 8
    for (int k = 0; k < Hq; ++k) s += hr[k] * c3w[p * Hq + k];
    out[idx] += s;
}

// ---------------------------------------------------------------- host orchestration
extern "C" void kernel_launch(void* const* d_in, const int* in_sizes, int n_in,
                              void* d_out, int out_size, void* d_ws, size_t ws_size,
                              hipStream_t stream) {
    (void)in_sizes; (void)n_in; (void)out_size; (void)ws_size;
    const float* src  = (const float*)d_in[0];
    const float* ne1  = (const float*)d_in[1];
    const float* temb = (const float*)d_in[2];
    const float* demb = (const float*)d_in[3];
    const float* c1w = (const float*)d_in[36]; const float* c1b = (const float*)d_in[37];
    const float* c2w = (const float*)d_in[38]; const float* c2b = (const float*)d_in[39];
    const float* c3w = (const float*)d_in[40]; const float* c3b = (const float*)d_in[41];
    float* out = (float*)d_out;

    float* w = (float*)d_ws;
    size_t off = 0;
    float* e1    = w + off; off += (size_t)Bq * Tq * Nq * Dq;
    float* state = w + off; off += (size_t)Bq * Nq * Hq;
    float* xin   = w + off; off += (size_t)Bq * NP * XP;
    float* nv    = w + off; off += (size_t)Bq * NP * DP;
    float* dinv  = w + off; off += (size_t)Bq * NP;
    float* yagg  = w + off; off += (size_t)Bq * NP * XP;
    float* zr    = w + off; off += (size_t)Bq * Nq * 128;
    float* hc    = w + off; off += (size_t)Bq * Nq * Hq;
    float* src2  = w + off; off += (size_t)Bq * Tq * Nq;

    // zero padded buffers once per launch (padding regions stay zero thereafter)
    fill_kernel<<<(Bq * NP * XP + 255) / 256, 256, 0, stream>>>(xin, Bq * NP * XP, 0.0f);
    fill_kernel<<<(Bq * NP * DP + 255) / 256, 256, 0, stream>>>(nv, Bq * NP * DP, 0.0f);
    fill_kernel<<<(Bq * NP + 255) / 256, 256, 0, stream>>>(dinv, Bq * NP, 0.0f);

    embed_kernel<<<(Bq * Tq * Nq + 255) / 256, 256, 0, stream>>>(src, ne1, temb, demb, e1);

    for (int enc = 0; enc < 2; ++enc) {
        const float* xsrc = (enc == 0) ? src : src2;
        int xstride = (enc == 0) ? 3 : 1;
        fill_kernel<<<(Bq * Nq * Hq + 255) / 256, 256, 0, stream>>>(state, Bq * Nq * Hq, 0.0f);
        for (int t = 0; t < Tq; ++t) {
            for (int which = 0; which < 2; ++which) {   // 0=gate, 1=update
                int base = 4 + enc * 16 + which * 8;
                const float* wp  = (const float*)d_in[base + 0];
                const float* bp  = (const float*)d_in[base + 1];
                const float* f1w = (const float*)d_in[base + 2];
                const float* f1b = (const float*)d_in[base + 3];
                const float* f2w = (const float*)d_in[base + 4];
                const float* f2b = (const float*)d_in[base + 5];
                const float* f3w = (const float*)d_in[base + 6];
                const float* f3b = (const float*)d_in[base + 7];
                build_xin_kernel<<<(Bq * Nq + 255) / 256, 256, 0, stream>>>(
                    xsrc, xstride, t, state, which ? zr : nullptr, xin);
                mlp_nodevec_kernel<<<(Bq * Nq + 255) / 256, 256, 0, stream>>>(
                    xin, e1, t, f1w, f1b, f2w, f2b, f3w, f3b, nv);
                dinv_kernel<<<dim3(Bq, (Nq + 255) / 256), 256, 0, stream>>>(nv, dinv);
                lap_agg_kernel<<<dim3(NTILES, Bq), 32, 0, stream>>>(nv, dinv, xin, yagg);
                int dout = which ? Hq : 2 * Hq;
                proj_kernel<<<dim3(Nq, dout / 64), 128, 0, stream>>>(
                    xin, yagg, ne1, wp, bp, dout, which, which ? hc : zr);
            }
            state_update_kernel<<<(Bq * Nq * Hq + 255) / 256, 256, 0, stream>>>(state, zr, hc);
        }
        if (enc == 0)
            endconv1_kernel<<<(Bq * Tq * Nq + 255) / 256, 256, 0, stream>>>(
                state, c1w, c1b, c2w, c2b, src, out, src2);
        else
            endconv2_kernel<<<(Bq * Tq * Nq + 255) / 256, 256, 0, stream>>>(state, c3w, c3b, out);
    }
}